// SRCapsNet_40553081209483
// MI455X (gfx1250) — compile-verified
//
#include <hip/hip_runtime.h>
#include <hip/hip_bf16.h>

// ---------------------------------------------------------------------------
// SR-CapsNet routing on MI455X (gfx1250).
// Memory-bound streaming of pose weights (660MB + 176MB f32) through
// V_WMMA_F32_16X16X4_F32 accumulator chains. ~4 FLOP/byte -> HBM-bound,
// so the design goal is: read each pose weight exactly once, wide loads,
// lots of independent waves, all heavy math in WMMA.
// ---------------------------------------------------------------------------

typedef float v2f __attribute__((ext_vector_type(2)));
typedef float v8f __attribute__((ext_vector_type(8)));

#define Bb  8
#define Ll  30
#define Dd  512
#define NSs 21
#define NIi 8
#define LG  10   // l-groups for pose_ws partials (3 l each)
#define LPER 3
#define SG  7    // s-groups for pose_si partials (3 s each)
#define SPER 3

// ---------------- workspace layout (floats) ----------------
// wc         : 0                  B*L*NS   = 5040
// aw_buf     : 5040               B*L      = 240
// weighted_c : 5280               B*NS     = 168
// a_slots    : 5448               B*NS     = 168
// wc2        : 5616               B*NS*NI  = 1344
// weighted_c2: 6960               B*NI     = 64
// maxidx(int): 7024               B        = 8
// u_slots    : 7040               B*NS*D   = 86016
// u_part     : 93056              LG*B*NS*D= 860160
// v_part     : 953216             SG*B*NI*D= 229376
// total ~ 4.73 MB
#define WS_WC      0
#define WS_AW      5040
#define WS_WCSUM   5280
#define WS_ASLOT   5448
#define WS_WC2     5616
#define WS_WC2SUM  6960
#define WS_MAXIDX  7024
#define WS_USLOTS  7040
#define WS_UPART   93056
#define WS_VPART   953216

// ===========================================================================
// K1: route words -> slots. One wave32 per (b,l).
// ===========================================================================
__global__ __launch_bounds__(32) void k_route_ws(
    const float* __restrict__ tf, const float* __restrict__ Wr,
    float* __restrict__ wc, float* __restrict__ aw_buf,
    float* __restrict__ slots_out)
{
    int b = blockIdx.x / Ll, l = blockIdx.x % Ll;
    int n = threadIdx.x;
    const float* x = tf + ((size_t)b * Ll + l) * Dd;

    // a_words = mean over D (cooperative strided sum + butterfly reduce)
    float aw = 0.f;
    for (int j = n; j < Dd; j += 32) aw += x[j];
    for (int off = 16; off; off >>= 1) aw += __shfl_xor(aw, off, 32);
    aw *= (1.0f / Dd);

    // per-capsule logit
    float dot = -3.4e38f;
    if (n < NSs) {
        const float* w = Wr + ((size_t)l * NSs + n) * Dd;
        float d = 0.f;
        for (int j = 0; j < Dd; ++j) d += w[j] * x[j];
        dot = d;
    }
    // softmax across lanes 0..20
    float m = dot;
    for (int off = 16; off; off >>= 1) m = fmaxf(m, __shfl_xor(m, off, 32));
    float e = (n < NSs) ? __expf(dot - m) : 0.f;
    float sum = e;
    for (int off = 16; off; off >>= 1) sum += __shfl_xor(sum, off, 32);
    float c = e / sum;
    if (n < NSs) wc[((size_t)b * Ll + l) * NSs + n] = c * aw;
    if (n == 0)  aw_buf[b * Ll + l] = aw;

    // argmax(c_ws) == argmax(logit); min-index tie break
    float bd = dot; int bi = (n < NSs) ? n : 0;
    for (int off = 16; off; off >>= 1) {
        float od = __shfl_xor(bd, off, 32);
        int   oi = __shfl_xor(bi, off, 32);
        if (od > bd || (od == bd && oi < bi)) { bd = od; bi = oi; }
    }
    if (n == 0) slots_out[b * Ll + l] = (float)bi;
}

// ===========================================================================
// K1b: weighted_c / a_slots (deterministic reduction over l). One wave per b.
// ===========================================================================
__global__ __launch_bounds__(32) void k_slot_stats(
    const float* __restrict__ wc, const float* __restrict__ aw_buf,
    float* __restrict__ weighted_c, float* __restrict__ a_slots)
{
    int b = blockIdx.x, n = threadIdx.x;
    float awsum = 0.f;
    for (int l = 0; l < Ll; ++l) awsum += aw_buf[b * Ll + l];
    if (n < NSs) {
        float s = 0.f;
        for (int l = 0; l < Ll; ++l) s += wc[((size_t)b * Ll + l) * NSs + n];
        weighted_c[b * NSs + n] = s;
        a_slots[b * NSs + n] = s / awsum;
    }
}

// ===========================================================================
// K2: pose words->slots, the 660MB stream.  D = A(8xK) * W^T via
// V_WMMA_F32_16X16X4_F32.  grid = (s=21, ktile=4, lgroup=10), block = 128.
// Each wave owns 32 output columns (two 16-wide WMMA tiles).
// A/B VGPR layout per ISA 16x4-f32: element[idx][k] -> lane idx+16*(k>>1),
// vgpr k&1  => each lane does one b64 load per operand per K-step.
// ===========================================================================
__global__ __launch_bounds__(128) void k_pose_ws(
    const float* __restrict__ tf, const float* __restrict__ Wp,
    const float* __restrict__ wc, float* __restrict__ u_part)
{
    int s     = blockIdx.x;
    int ktile = blockIdx.y;
    int g     = blockIdx.z;
    int wave  = threadIdx.x >> 5;
    int lane  = threadIdx.x & 31;
    int n0    = ktile * 128 + wave * 32;
    int mrow  = lane & 15;
    int kpair = lane >> 4;

    v8f acc0 = {}, acc1 = {};

    for (int li = 0; li < LPER; ++li) {
        int l = g * LPER + li;
        const float* Abase  = tf + ((size_t)(mrow & 7) * Ll + l) * Dd;
        const float* B0base = Wp + ((((size_t)l * NSs + s) * Dd) + n0 + mrow) * Dd;
        const float* B1base = B0base + (size_t)16 * Dd;

        v8f c0 = {}, c1 = {};
        for (int j = 0; j < Dd; j += 4) {
            int jo = j + 2 * kpair;
            v2f a  = *(const v2f*)(Abase  + jo);
            v2f b0 = *(const v2f*)(B0base + jo);
            v2f b1 = *(const v2f*)(B1base + jo);
            c0 = __builtin_amdgcn_wmma_f32_16x16x4_f32(false, a, false, b0,
                                                       (short)0, c0, false, false);
            c1 = __builtin_amdgcn_wmma_f32_16x16x4_f32(false, a, false, b1,
                                                       (short)0, c1, false, false);
        }
        // C layout: vgpr v <-> batch row b=v (lanes 0-15).  wc scale is a
        // uniform scalar per vgpr -> s_load + v_fma.
#pragma unroll
        for (int v = 0; v < 8; ++v) {
            float w = wc[((size_t)v * Ll + l) * NSs + s];
            acc0[v] += w * c0[v];
            acc1[v] += w * c1[v];
        }
    }
    if (lane < 16) {
#pragma unroll
        for (int v = 0; v < 8; ++v) {
            size_t base = (((size_t)g * Bb + v) * NSs + s) * Dd;
            u_part[base + n0 + lane]      = acc0[v];
            u_part[base + n0 + 16 + lane] = acc1[v];
        }
    }
}

// ===========================================================================
// K3: u_slots = sum_g u_part / weighted_c
// ===========================================================================
__global__ __launch_bounds__(128) void k_reduce_us(
    const float* __restrict__ u_part, const float* __restrict__ weighted_c,
    float* __restrict__ u_slots)
{
    int bs = blockIdx.x;               // b*NS + s
    int b = bs / NSs, s = bs % NSs;
    float invw = 1.f / weighted_c[bs];
    for (int k = threadIdx.x; k < Dd; k += blockDim.x) {
        float acc = 0.f;
        for (int g = 0; g < LG; ++g)
            acc += u_part[(((size_t)g * Bb + b) * NSs + s) * Dd + k];
        u_slots[(size_t)bs * Dd + k] = acc * invw;
    }
}

// ===========================================================================
// K4: route slots -> intents. One wave per (b,s).
// ===========================================================================
__global__ __launch_bounds__(32) void k_route_si(
    const float* __restrict__ us, const float* __restrict__ Wr2,
    const float* __restrict__ a_slots, float* __restrict__ wc2)
{
    int b = blockIdx.x / NSs, s = blockIdx.x % NSs;
    int i = threadIdx.x;
    const float* x = us + ((size_t)b * NSs + s) * Dd;
    float dot = -3.4e38f;
    if (i < NIi) {
        const float* w = Wr2 + ((size_t)s * NIi + i) * Dd;
        float d = 0.f;
        for (int j = 0; j < Dd; ++j) d += w[j] * x[j];
        dot = d;
    }
    float m = dot;
    for (int off = 16; off; off >>= 1) m = fmaxf(m, __shfl_xor(m, off, 32));
    float e = (i < NIi) ? __expf(dot - m) : 0.f;
    float sum = e;
    for (int off = 16; off; off >>= 1) sum += __shfl_xor(sum, off, 32);
    if (i < NIi)
        wc2[((size_t)b * NSs + s) * NIi + i] = (e / sum) * a_slots[b * NSs + s];
}

// ===========================================================================
// K4b: weighted_c2, a_intents argmax. One wave per b.
// ===========================================================================
__global__ __launch_bounds__(32) void k_int_stats(
    const float* __restrict__ wc2, const float* __restrict__ a_slots,
    float* __restrict__ weighted_c2, int* __restrict__ maxidx)
{
    int b = blockIdx.x, i = threadIdx.x;
    float asum = 0.f;
    for (int s = 0; s < NSs; ++s) asum += a_slots[b * NSs + s];
    float wsum = -3.4e38f;
    if (i < NIi) {
        float t = 0.f;
        for (int s = 0; s < NSs; ++s) t += wc2[((size_t)b * NSs + s) * NIi + i];
        weighted_c2[b * NIi + i] = t;
        wsum = t / asum;               // a_intents
    }
    float bd = wsum; int bi = (i < NIi) ? i : 0;
    for (int off = 16; off; off >>= 1) {
        float od = __shfl_xor(bd, off, 32);
        int   oi = __shfl_xor(bi, off, 32);
        if (od > bd || (od == bd && oi < bi)) { bd = od; bi = oi; }
    }
    if (i == 0) maxidx[b] = bi;
}

// ===========================================================================
// K5: pose slots->intents, the 176MB stream.  grid = (i=8, ktile=4, sg=7).
// ===========================================================================
__global__ __launch_bounds__(128) void k_pose_si(
    const float* __restrict__ us, const float* __restrict__ Wp2,
    const float* __restrict__ wc2, float* __restrict__ v_part)
{
    int i     = blockIdx.x;
    int ktile = blockIdx.y;
    int g     = blockIdx.z;
    int wave  = threadIdx.x >> 5;
    int lane  = threadIdx.x & 31;
    int n0    = ktile * 128 + wave * 32;
    int mrow  = lane & 15;
    int kpair = lane >> 4;

    v8f acc0 = {}, acc1 = {};

    for (int si = 0; si < SPER; ++si) {
        int s = g * SPER + si;
        const float* Abase  = us + ((size_t)(mrow & 7) * NSs + s) * Dd;
        const float* B0base = Wp2 + ((((size_t)s * NIi + i) * Dd) + n0 + mrow) * Dd;
        const float* B1base = B0base + (size_t)16 * Dd;

        v8f c0 = {}, c1 = {};
        for (int j = 0; j < Dd; j += 4) {
            int jo = j + 2 * kpair;
            v2f a  = *(const v2f*)(Abase  + jo);
            v2f b0 = *(const v2f*)(B0base + jo);
            v2f b1 = *(const v2f*)(B1base + jo);
            c0 = __builtin_amdgcn_wmma_f32_16x16x4_f32(false, a, false, b0,
                                                       (short)0, c0, false, false);
            c1 = __builtin_amdgcn_wmma_f32_16x16x4_f32(false, a, false, b1,
                                                       (short)0, c1, false, false);
        }
#pragma unroll
        for (int v = 0; v < 8; ++v) {
            float w = wc2[((size_t)v * NSs + s) * NIi + i];
            acc0[v] += w * c0[v];
            acc1[v] += w * c1[v];
        }
    }
    if (lane < 16) {
#pragma unroll
        for (int v = 0; v < 8; ++v) {
            size_t base = (((size_t)g * Bb + v) * NIi + i) * Dd;
            v_part[base + n0 + lane]      = acc0[v];
            v_part[base + n0 + 16 + lane] = acc1[v];
        }
    }
}

// ===========================================================================
// K6: pick argmax intent, reduce its partials, add cls residual.
// ===========================================================================
__global__ __launch_bounds__(128) void k_finish(
    const float* __restrict__ v_part, const float* __restrict__ weighted_c2,
    const int* __restrict__ maxidx, const float* __restrict__ cls,
    float* __restrict__ out)
{
    int b = blockIdx.x;
    int istar = maxidx[b];
    float invw = 1.f / weighted_c2[b * NIi + istar];
    for (int k = threadIdx.x; k < Dd; k += blockDim.x) {
        float acc = 0.f;
        for (int g = 0; g < SG; ++g)
            acc += v_part[(((size_t)g * Bb + b) * NIi + istar) * Dd + k];
        out[(size_t)b * Dd + k] = cls[(size_t)b * Dd + k] + acc * invw;
    }
}

// ===========================================================================
extern "C" void kernel_launch(void* const* d_in, const int* in_sizes, int n_in,
                              void* d_out, int out_size, void* d_ws, size_t ws_size,
                              hipStream_t stream)
{
    const float* tf   = (const float*)d_in[0];   // (B,L,D)
    const float* cls  = (const float*)d_in[1];   // (B,D)
    const float* Wrws = (const float*)d_in[2];   // (L,NS,D)
    const float* Wpws = (const float*)d_in[3];   // (L,NS,D,D)
    const float* Wrsi = (const float*)d_in[4];   // (NS,NI,D)
    const float* Wpsi = (const float*)d_in[5];   // (NS,NI,D,D)

    float* F = (float*)d_ws;
    float* wc        = F + WS_WC;
    float* aw_buf    = F + WS_AW;
    float* wcsum     = F + WS_WCSUM;
    float* a_slots   = F + WS_ASLOT;
    float* wc2       = F + WS_WC2;
    float* wc2sum    = F + WS_WC2SUM;
    int*   maxidx    = (int*)(F + WS_MAXIDX);
    float* u_slots   = F + WS_USLOTS;
    float* u_part    = F + WS_UPART;
    float* v_part    = F + WS_VPART;

    float* slots_out   = (float*)d_out;             // B*L
    float* intents_out = (float*)d_out + Bb * Ll;   // B*D

    k_route_ws  <<<dim3(Bb * Ll), dim3(32), 0, stream>>>(tf, Wrws, wc, aw_buf, slots_out);
    k_slot_stats<<<dim3(Bb),      dim3(32), 0, stream>>>(wc, aw_buf, wcsum, a_slots);
    k_pose_ws   <<<dim3(NSs, 4, LG), dim3(128), 0, stream>>>(tf, Wpws, wc, u_part);
    k_reduce_us <<<dim3(Bb * NSs), dim3(128), 0, stream>>>(u_part, wcsum, u_slots);
    k_route_si  <<<dim3(Bb * NSs), dim3(32), 0, stream>>>(u_slots, Wrsi, a_slots, wc2);
    k_int_stats <<<dim3(Bb),       dim3(32), 0, stream>>>(wc2, a_slots, wc2sum, maxidx);
    k_pose_si   <<<dim3(NIi, 4, SG), dim3(128), 0, stream>>>(u_slots, Wpsi, wc2, v_part);
    k_finish    <<<dim3(Bb),       dim3(128), 0, stream>>>(v_part, wc2sum, maxidx, cls, intents_out);
}